// TinyAttention_83760452206744
// MI455X (gfx1250) — compile-verified
//
#include <hip/hip_runtime.h>
#include <math.h>

// Problem constants (reference: B=4, T=2048, C=1024, H=16, hd=64)
#define H_  16
#define T_  2048
#define C_  1024
#define HD  64

typedef __attribute__((ext_vector_type(2))) float v2f;
typedef __attribute__((ext_vector_type(8))) float v8f;

__device__ __forceinline__ v8f splat8(float x) {
  v8f r;
#pragma unroll
  for (int i = 0; i < 8; ++i) r[i] = x;
  return r;
}

// ---------------------------------------------------------------------------
// Weight transpose into pair-interleaved layout:
//   in  : [rows=N][cols=K]   (torch Linear weight, w[n][k])
//   out : [K/2][N][2]        out[((k>>1)*N + n)*2 + (k&1)] = w[n][k]
// A lane's WMMA B-fragment (rows k,k+1 at column n) is then one b64 load.
// ---------------------------------------------------------------------------
__global__ __launch_bounds__(256) void tr_pair_kernel(const float* __restrict__ in,
                                                      float* __restrict__ out,
                                                      int rows, int cols) {
  __shared__ float tile[32][33];
  int c  = blockIdx.x * 32 + threadIdx.x;   // k
  int r0 = blockIdx.y * 32 + threadIdx.y;   // n
#pragma unroll
  for (int i = 0; i < 32; i += 8) {
    int r = r0 + i;
    if (r < rows && c < cols) tile[threadIdx.y + i][threadIdx.x] = in[(size_t)r * cols + c];
  }
  __syncthreads();
  int oc  = blockIdx.y * 32 + threadIdx.x;  // n (output col)
  int or0 = blockIdx.x * 32 + threadIdx.y;  // k (output row)
#pragma unroll
  for (int i = 0; i < 32; i += 8) {
    int k = or0 + i;
    if (k < cols && oc < rows)
      out[((size_t)(k >> 1) * rows + oc) * 2 + (k & 1)] = tile[threadIdx.x][threadIdx.y + i];
  }
}

// ---------------------------------------------------------------------------
// One wave computes a 32(M) x 64(N) strip with fp32 WMMA.
// Per k-step: 2 A b64 loads + 4 B b64 loads feed 8 WMMAs.
// BTP is pair-interleaved transposed weight [K/2][N][2].
// ---------------------------------------------------------------------------
__device__ __forceinline__ void gemm_strip2_f32(const float* __restrict__ A,
                                                const float* __restrict__ BTP,
                                                int K, int N, int m0, int n0,
                                                int lane,
                                                const float* __restrict__ bias,
                                                v8f acc[2][4]) {
  const int half = lane >> 4;
  const int l16  = lane & 15;
#pragma unroll
  for (int i = 0; i < 2; ++i)
#pragma unroll
    for (int j = 0; j < 4; ++j) acc[i][j] = splat8(bias[n0 + 16 * j + l16]);

  const float* arow0 = A + (size_t)(m0 + l16) * K + 2 * half;
  const float* arow1 = arow0 + (size_t)16 * K;
  for (int k = 0; k < K; k += 4) {
    v2f a0 = *(const v2f*)(arow0 + k);
    v2f a1 = *(const v2f*)(arow1 + k);
    // paired-k row: (k + 2*half)/2 = k/2 + half
    const float* bk = BTP + (size_t)((k >> 1) + half) * (2 * N);
#pragma unroll
    for (int j = 0; j < 4; ++j) {
      v2f b = *(const v2f*)(bk + 2 * (n0 + 16 * j + l16));
      acc[0][j] = __builtin_amdgcn_wmma_f32_16x16x4_f32(false, a0, false, b,
                                                        (short)0, acc[0][j], false, false);
      acc[1][j] = __builtin_amdgcn_wmma_f32_16x16x4_f32(false, a1, false, b,
                                                        (short)0, acc[1][j], false, false);
    }
  }
}

// ---------------------------------------------------------------------------
// QKV projection: qkv = x @ qkv_w^T + qkv_b, scatter-stored as
//   Q   [B,H,T,hd]            (A-frag b64 loads in attention)
//   KTP [B,H,hd/2,T,2]        pair-interleaved along hd -> S B-frags are b64
//   VP  [B,H,T/2,hd,2]        pair-interleaved along T  -> PV B-frags are b64
// ---------------------------------------------------------------------------
__global__ __launch_bounds__(128) void qkv_kernel(const float* __restrict__ x,
                                                  const float* __restrict__ wTP,
                                                  const float* __restrict__ bias,
                                                  float* __restrict__ qv,
                                                  float* __restrict__ kTp,
                                                  float* __restrict__ vp) {
  const int wave = (int)((blockIdx.x * blockDim.x + threadIdx.x) >> 5);
  const int lane = threadIdx.x & 31;
  const int NSTRIP = (3 * C_) / 64;  // 48
  const int m0 = (wave / NSTRIP) * 32;
  const int n0 = (wave % NSTRIP) * 64;

  v8f acc[2][4];
  gemm_strip2_f32(x, wTP, C_, 3 * C_, m0, n0, lane, bias, acc);

  const int half = lane >> 4, l16 = lane & 15;
#pragma unroll
  for (int i = 0; i < 2; ++i) {
#pragma unroll
    for (int j = 0; j < 4; ++j) {
      const int nn = n0 + 16 * j + l16;
#pragma unroll
      for (int r = 0; r < 8; ++r) {
        const int mg = m0 + 16 * i + r + 8 * half;
        const int bb = mg >> 11;        // / T_
        const int t  = mg & (T_ - 1);
        const float val = acc[i][j][r];
        if (nn < C_) {
          const int h = nn >> 6, d = nn & 63;
          qv[((size_t)(bb * H_ + h) * T_ + t) * HD + d] = val;
        } else if (nn < 2 * C_) {
          const int n2 = nn - C_;
          const int h = n2 >> 6, d = n2 & 63;
          kTp[(((size_t)(bb * H_ + h) * (HD / 2) + (d >> 1)) * T_ + t) * 2 + (d & 1)] = val;
        } else {
          const int n2 = nn - 2 * C_;
          const int h = n2 >> 6, d = n2 & 63;
          vp[(((size_t)(bb * H_ + h) * (T_ / 2) + (t >> 1)) * HD + d) * 2 + (t & 1)] = val;
        }
      }
    }
  }
}

// ---------------------------------------------------------------------------
// Flash-style attention: one wave owns 16 query rows of one (b,h); loops over
// kv tiles of 16, doing S=QK^T (16 WMMAs), online softmax, and O += P.V
// (16 WMMAs) with a per-wave LDS transpose of P into A-fragment layout.
// All global B-fragments are single b64 loads via pair-interleaved K/V.
// Output written as [B,T,H*hd] so the proj GEMM reads it row-major.
// ---------------------------------------------------------------------------
__global__ __launch_bounds__(128) void attn_kernel(const float* __restrict__ q,
                                                   const float* __restrict__ kTp,
                                                   const float* __restrict__ vpI,
                                                   float* __restrict__ aout) {
  __shared__ float pl[4][16][18];   // pad 18 -> 8B-aligned b64 A-frag reads
  const int waveid = threadIdx.x >> 5;
  const int lane   = threadIdx.x & 31;
  const int wave   = blockIdx.x * 4 + waveid;
  const int bh     = wave >> 7;          // / (T_/16)
  const int qtile  = wave & 127;
  const int b = bh / H_, h = bh % H_;
  const int qt0 = qtile * 16;
  const float* qb = q   + (size_t)bh * T_ * HD;
  const float* kb = kTp + (size_t)bh * HD * T_;   // [hd/2][T][2]
  const float* vb = vpI + (size_t)bh * T_ * HD;   // [T/2][hd][2]
  const int half = lane >> 4, l16 = lane & 15;
  const float scale = 0.125f;            // hd^-0.5

  // Preload Q A-fragments for all 16 k-steps (hd=64)
  v2f qa[16];
#pragma unroll
  for (int kk = 0; kk < 16; ++kk)
    qa[kk] = *(const v2f*)(qb + (size_t)(qt0 + l16) * HD + 4 * kk + 2 * half);

  v8f oacc[4];
#pragma unroll
  for (int j = 0; j < 4; ++j) oacc[j] = splat8(0.f);
  float mrow[8], lrow[8];
#pragma unroll
  for (int r = 0; r < 8; ++r) { mrow[r] = -INFINITY; lrow[r] = 0.f; }

  float (*P)[18] = pl[waveid];

  for (int kv0 = 0; kv0 < T_; kv0 += 16) {
    // S = Q K^T (16x16 tile); B-frag = b64 from pair-interleaved K^T
    v8f s = splat8(0.f);
#pragma unroll
    for (int kk = 0; kk < 16; ++kk) {
      v2f bfr = *(const v2f*)(kb + ((size_t)(2 * kk + half) * T_ + kv0 + l16) * 2);
      s = __builtin_amdgcn_wmma_f32_16x16x4_f32(false, qa[kk], false, bfr,
                                                (short)0, s, false, false);
    }
    // Online softmax: each D-layout VGPR r holds one row across a 16-lane group
    float alpha[8];
#pragma unroll
    for (int r = 0; r < 8; ++r) {
      float sv = s[r] * scale;
      float mx = sv;
      mx = fmaxf(mx, __shfl_xor(mx, 1));
      mx = fmaxf(mx, __shfl_xor(mx, 2));
      mx = fmaxf(mx, __shfl_xor(mx, 4));
      mx = fmaxf(mx, __shfl_xor(mx, 8));
      const float mnew = fmaxf(mrow[r], mx);
      alpha[r] = __expf(mrow[r] - mnew);
      mrow[r]  = mnew;
      const float p = __expf(sv - mnew);
      s[r] = p;
      float sum = p;
      sum += __shfl_xor(sum, 1);
      sum += __shfl_xor(sum, 2);
      sum += __shfl_xor(sum, 4);
      sum += __shfl_xor(sum, 8);
      lrow[r] = lrow[r] * alpha[r] + sum;
    }
#pragma unroll
    for (int j = 0; j < 4; ++j)
#pragma unroll
      for (int r = 0; r < 8; ++r) oacc[j][r] *= alpha[r];

    // Transpose P (D layout -> A layout) via per-wave LDS tile; DS ops are
    // in-order within a wave, no barrier needed.
#pragma unroll
    for (int r = 0; r < 8; ++r) P[r + 8 * half][l16] = s[r];

    // O += P . V ; B-frag = b64 from pair-interleaved V
#pragma unroll
    for (int kk = 0; kk < 4; ++kk) {
      v2f pa = *(const v2f*)&P[l16][4 * kk + 2 * half];
      const int tp = (kv0 + 4 * kk + 2 * half) >> 1;   // even numerator
#pragma unroll
      for (int j = 0; j < 4; ++j) {
        v2f bv = *(const v2f*)(vb + ((size_t)tp * HD + 16 * j + l16) * 2);
        oacc[j] = __builtin_amdgcn_wmma_f32_16x16x4_f32(false, pa, false, bv,
                                                        (short)0, oacc[j], false, false);
      }
    }
  }

  float inv[8];
#pragma unroll
  for (int r = 0; r < 8; ++r) inv[r] = 1.0f / lrow[r];
#pragma unroll
  for (int j = 0; j < 4; ++j)
#pragma unroll
    for (int r = 0; r < 8; ++r) {
      const int t = qt0 + r + 8 * half;
      aout[((size_t)b * T_ + t) * C_ + h * HD + 16 * j + l16] = oacc[j][r] * inv[r];
    }
}

// ---------------------------------------------------------------------------
// Output projection: out = attn @ proj_w^T + proj_b
// ---------------------------------------------------------------------------
__global__ __launch_bounds__(128) void proj_kernel(const float* __restrict__ Ain,
                                                   const float* __restrict__ wTP,
                                                   const float* __restrict__ bias,
                                                   float* __restrict__ out) {
  const int wave = (int)((blockIdx.x * blockDim.x + threadIdx.x) >> 5);
  const int lane = threadIdx.x & 31;
  const int NSTRIP = C_ / 64;  // 16
  const int m0 = (wave / NSTRIP) * 32;
  const int n0 = (wave % NSTRIP) * 64;

  v8f acc[2][4];
  gemm_strip2_f32(Ain, wTP, C_, C_, m0, n0, lane, bias, acc);

  const int half = lane >> 4, l16 = lane & 15;
#pragma unroll
  for (int i = 0; i < 2; ++i)
#pragma unroll
    for (int j = 0; j < 4; ++j)
#pragma unroll
      for (int r = 0; r < 8; ++r) {
        const int mg = m0 + 16 * i + r + 8 * half;
        out[(size_t)mg * C_ + n0 + 16 * j + l16] = acc[i][j][r];
      }
}

// ---------------------------------------------------------------------------
extern "C" void kernel_launch(void* const* d_in, const int* in_sizes, int n_in,
                              void* d_out, int out_size, void* d_ws, size_t ws_size,
                              hipStream_t stream) {
  const float* x      = (const float*)d_in[0];
  const float* qkv_w  = (const float*)d_in[1];
  const float* qkv_b  = (const float*)d_in[2];
  const float* proj_w = (const float*)d_in[3];
  const float* proj_b = (const float*)d_in[4];
  float* out = (float*)d_out;

  const int B = in_sizes[0] / (T_ * C_);   // 4
  const int M = B * T_;                    // 8192

  // Workspace layout (floats)
  float* ws      = (float*)d_ws;
  float* qkv_wTP  = ws;                                   // [C/2][3C][2]
  float* proj_wTP = qkv_wTP + (size_t)3 * C_ * C_;        // [C/2][C][2]
  float* Q        = proj_wTP + (size_t)C_ * C_;           // [B,H,T,hd]
  float* KTP      = Q       + (size_t)B * H_ * T_ * HD;   // [B,H,hd/2,T,2]
  float* VP       = KTP     + (size_t)B * H_ * T_ * HD;   // [B,H,T/2,hd,2]
  float* AO       = VP      + (size_t)B * H_ * T_ * HD;   // [B,T,C]

  dim3 trb(32, 8);
  tr_pair_kernel<<<dim3(C_ / 32, (3 * C_) / 32), trb, 0, stream>>>(qkv_w, qkv_wTP, 3 * C_, C_);
  tr_pair_kernel<<<dim3(C_ / 32, C_ / 32),       trb, 0, stream>>>(proj_w, proj_wTP, C_, C_);

  // QKV GEMM: (M/32) m-tiles x (3C/64) n-strips waves, 4 waves/block
  {
    const int waves = (M / 32) * ((3 * C_) / 64);
    qkv_kernel<<<waves / 4, 128, 0, stream>>>(x, qkv_wTP, qkv_b, Q, KTP, VP);
  }
  // Attention: one wave per 16 query rows per (b,h)
  {
    const int waves = B * H_ * (T_ / 16);
    attn_kernel<<<waves / 4, 128, 0, stream>>>(Q, KTP, VP, AO);
  }
  // Projection GEMM
  {
    const int waves = (M / 32) * (C_ / 64);
    proj_kernel<<<waves / 4, 128, 0, stream>>>(AO, proj_wTP, proj_b, out);
  }
}